// GATModel_61272003445042
// MI455X (gfx1250) — compile-verified
//
#include <hip/hip_runtime.h>
#include <hip/hip_bf16.h>

typedef __attribute__((ext_vector_type(2))) float v2f;
typedef __attribute__((ext_vector_type(8))) float v8f;

#define N_NODES 100000
#define N_EDGES 1600000
#define N_GRAPHS 2000

// ---------------------------------------------------------------------------
// zero helper
// ---------------------------------------------------------------------------
__global__ void k_zero(float* __restrict__ p, int n) {
    int i = blockIdx.x * blockDim.x + threadIdx.x;
    if (i < n) p[i] = 0.f;
}

// ---------------------------------------------------------------------------
// node embedding: h[n][d] = sum_c node_tab[c][X[n][c]][d]   (d in 0..15)
// ---------------------------------------------------------------------------
__global__ void k_encode_nodes(const int* __restrict__ X, const float* __restrict__ tab,
                               float* __restrict__ h, int N) {
    int gid = blockIdx.x * blockDim.x + threadIdx.x;
    int n = gid >> 4, d = gid & 15;
    if (n >= N) return;
    float s = 0.f;
#pragma unroll
    for (int c = 0; c < 9; ++c) {
        int v = X[n * 9 + c];
        s += tab[(c * 119 + v) * 16 + d];
    }
    h[(size_t)n * 16 + d] = s;
}

// ---------------------------------------------------------------------------
// edge embedding: ea[e][0..1] = sum_c edge_tab[c][Ed[e][c]][0..1]
// ---------------------------------------------------------------------------
__global__ void k_encode_edges(const int* __restrict__ Ed, const float* __restrict__ tab,
                               float* __restrict__ ea, int E) {
    int e = blockIdx.x * blockDim.x + threadIdx.x;
    if (e >= E) return;
    float s0 = 0.f, s1 = 0.f;
#pragma unroll
    for (int c = 0; c < 3; ++c) {
        int v = Ed[e * 3 + c];
        const float* t = tab + (c * 6 + v) * 2;
        s0 += t[0];
        s1 += t[1];
    }
    ea[(size_t)e * 2 + 0] = s0;
    ea[(size_t)e * 2 + 1] = s1;
}

// ---------------------------------------------------------------------------
// per-dst edge stats for self-loop fill_value='mean'
// ---------------------------------------------------------------------------
__global__ void k_edge_stats(const int* __restrict__ dst, const float* __restrict__ ea,
                             float* __restrict__ cnt, float* __restrict__ easum, int E) {
    int e = blockIdx.x * blockDim.x + threadIdx.x;
    if (e >= E) return;
    int d = dst[e];
    atomicAdd(&cnt[d], 1.f);
    atomicAdd(&easum[(size_t)d * 2 + 0], ea[(size_t)e * 2 + 0]);
    atomicAdd(&easum[(size_t)d * 2 + 1], ea[(size_t)e * 2 + 1]);
}

// ---------------------------------------------------------------------------
// fold (ea @ We) @ ae  ->  ea . (We @ ae) :  aecoef[j] = sum_k We[j][k]*ae[k]
// ---------------------------------------------------------------------------
__global__ void k_aecoef(const float* __restrict__ We, const float* __restrict__ ae,
                         float* __restrict__ out) {
    int j = threadIdx.x;
    if (j >= 2) return;
    float s = 0.f;
#pragma unroll
    for (int k = 0; k < 32; ++k) s += We[j * 32 + k] * ae[k];
    out[j] = s;
}

// ---------------------------------------------------------------------------
// WMMA GEMM: xw[N,32] = x[N,D] @ W[D,32] with V_WMMA_F32_16X16X4_F32.
// 256-thread block = 8 wave32s; each wave owns one 16-row M tile, two 16x16
// accumulators cover N=32. K stepped by 4 per the f32 A/B VGPR layouts.
// ---------------------------------------------------------------------------
__global__ void k_gemm_wmma(const float* __restrict__ x, const float* __restrict__ W,
                            float* __restrict__ xw, int N, int D) {
    __shared__ float sW[32 * 32];
    int tid = threadIdx.x;
    for (int i = tid; i < D * 32; i += blockDim.x) sW[i] = W[i];
    __syncthreads();

    int wave = tid >> 5;
    int lane = tid & 31;
    int m0 = (blockIdx.x * 8 + wave) * 16;
    if (m0 >= N) return;

    int half = lane >> 4;   // 0: K-pair {0,1}; 1: K-pair {2,3}
    int ln = lane & 15;
    int row = m0 + ln;
    if (row >= N) row = N - 1;            // clamp so EXEC stays all-ones for WMMA
    const float* xr = x + (size_t)row * D;

    v8f c0 = {};
    v8f c1 = {};
    for (int k = 0; k < D; k += 4) {
        v2f a, b0, b1;
        // A 16x4 f32: lanes 0-15 hold K=k..k+1, lanes 16-31 hold K=k+2..k+3
        a.x = xr[k + 2 * half + 0];
        a.y = xr[k + 2 * half + 1];
        // B 4x16 f32: lane = column N, vgpr = K row (upper lanes shift K by 2)
        b0.x = sW[(k + 2 * half + 0) * 32 + ln];
        b0.y = sW[(k + 2 * half + 1) * 32 + ln];
        b1.x = sW[(k + 2 * half + 0) * 32 + 16 + ln];
        b1.y = sW[(k + 2 * half + 1) * 32 + 16 + ln];
        c0 = __builtin_amdgcn_wmma_f32_16x16x4_f32(false, a, false, b0, (short)0, c0, false, false);
        c1 = __builtin_amdgcn_wmma_f32_16x16x4_f32(false, a, false, b1, (short)0, c1, false, false);
    }
    // C/D 16x16 f32: vgpr i -> M = i + 8*half, lane&15 -> N
#pragma unroll
    for (int i = 0; i < 8; ++i) {
        int r = m0 + i + half * 8;
        if (r < N) {
            xw[(size_t)r * 32 + ln]      = c0[i];
            xw[(size_t)r * 32 + 16 + ln] = c1[i];
        }
    }
}

// ---------------------------------------------------------------------------
// per-node prep for one GAT layer:
//  a_s, a_d dots; self-loop alpha (leaky relu) -> lval and maxbuf init;
//  den = 0; out[n][:] = bias
// ---------------------------------------------------------------------------
__global__ void k_node_prep(const float* __restrict__ xw, const float* __restrict__ as_,
                            const float* __restrict__ ad_, const float* __restrict__ cnt,
                            const float* __restrict__ easum, const float* __restrict__ aecoef,
                            const float* __restrict__ bias, float* __restrict__ a_s,
                            float* __restrict__ a_d, float* __restrict__ lval,
                            float* __restrict__ maxbuf, float* __restrict__ den,
                            float* __restrict__ out, int N) {
    int n = blockIdx.x * blockDim.x + threadIdx.x;
    if (n >= N) return;
    const float* xr = xw + (size_t)n * 32;
    float ss = 0.f, sd = 0.f;
#pragma unroll
    for (int k = 0; k < 32; ++k) {
        float v = xr[k];
        ss += v * as_[k];
        sd += v * ad_[k];
    }
    a_s[n] = ss;
    a_d[n] = sd;
    float c = cnt[n];
    if (c < 1.f) c = 1.f;
    float aeL = (easum[(size_t)n * 2 + 0] / c) * aecoef[0] +
                (easum[(size_t)n * 2 + 1] / c) * aecoef[1];
    float al = ss + sd + aeL;
    al = (al >= 0.f) ? al : 0.2f * al;
    lval[n] = al;
    maxbuf[n] = al;       // self-loop always present -> valid segment max seed
    den[n] = 0.f;
    float* o = out + (size_t)n * 32;
#pragma unroll
    for (int k = 0; k < 32; ++k) o[k] = bias[k];
}

// ---------------------------------------------------------------------------
// pass A: edge alpha + segment max (float atomicMax -> global_atomic_max_num_f32)
// ---------------------------------------------------------------------------
__global__ void k_edge_alpha(const int* __restrict__ src, const int* __restrict__ dst,
                             const float* __restrict__ ea, const float* __restrict__ aecoef,
                             const float* __restrict__ a_s, const float* __restrict__ a_d,
                             float* __restrict__ edgeval, float* __restrict__ maxbuf, int E) {
    int e = blockIdx.x * blockDim.x + threadIdx.x;
    if (e >= E) return;
    float aeE = ea[(size_t)e * 2 + 0] * aecoef[0] + ea[(size_t)e * 2 + 1] * aecoef[1];
    int d = dst[e];
    float al = a_s[src[e]] + a_d[d] + aeE;
    al = (al >= 0.f) ? al : 0.2f * al;
    edgeval[e] = al;
    atomicMax(&maxbuf[d], al);
}

// ---------------------------------------------------------------------------
// pass B: exp(alpha - max) and segment denominator (edges + self-loops fused)
// ---------------------------------------------------------------------------
__global__ void k_edge_exp(const int* __restrict__ dst, float* __restrict__ edgeval,
                           const float* __restrict__ maxbuf, float* __restrict__ den,
                           float* __restrict__ lval, int E, int N) {
    int i = blockIdx.x * blockDim.x + threadIdx.x;
    if (i < E) {
        int d = dst[i];
        float ex = __expf(edgeval[i] - maxbuf[d]);
        edgeval[i] = ex;
        atomicAdd(&den[d], ex);
    } else if (i < E + N) {
        int n = i - E;
        float ex = __expf(lval[n] - maxbuf[n]);
        lval[n] = ex;
        atomicAdd(&den[n], ex);
    }
}

// ---------------------------------------------------------------------------
// pass C: out[dst] += (ex/den[dst]) * xw[src]; one wave per edge, lane = channel
// (coalesced gathers of xw and coalesced atomic adds into out)
// ---------------------------------------------------------------------------
__global__ void k_aggregate(const int* __restrict__ src, const int* __restrict__ dst,
                            const float* __restrict__ edgeval, const float* __restrict__ den,
                            const float* __restrict__ xw, float* __restrict__ out,
                            const float* __restrict__ lval, int E, int N) {
    long long t = (long long)blockIdx.x * blockDim.x + threadIdx.x;
    long long i = t >> 5;
    int c = (int)(t & 31);
    if (i < E) {
        int s = src[i], d = dst[i];
        float w = edgeval[i] / den[d];
        atomicAdd(&out[(size_t)d * 32 + c], w * xw[(size_t)s * 32 + c]);
    } else if (i < (long long)E + N) {
        int n = (int)(i - E);
        float w = lval[n] / den[n];
        atomicAdd(&out[(size_t)n * 32 + c], w * xw[(size_t)n * 32 + c]);
    }
}

// ---------------------------------------------------------------------------
// fused pooling: s_n = dot([out1|out2|out3][n], lin_w); scalar atomic per node
// ---------------------------------------------------------------------------
__global__ void k_pool(const float* __restrict__ o1, const float* __restrict__ o2,
                       const float* __restrict__ o3, const float* __restrict__ lw,
                       const int* __restrict__ batch, float* __restrict__ gsum,
                       float* __restrict__ gcnt, int N) {
    int n = blockIdx.x * blockDim.x + threadIdx.x;
    if (n >= N) return;
    const float* a = o1 + (size_t)n * 32;
    const float* b = o2 + (size_t)n * 32;
    const float* c = o3 + (size_t)n * 32;
    float s = 0.f;
#pragma unroll
    for (int k = 0; k < 32; ++k) s += a[k] * lw[k];
#pragma unroll
    for (int k = 0; k < 32; ++k) s += b[k] * lw[32 + k];
#pragma unroll
    for (int k = 0; k < 32; ++k) s += c[k] * lw[64 + k];
    int g = batch[n];
    atomicAdd(&gsum[g], s);
    atomicAdd(&gcnt[g], 1.f);
}

__global__ void k_final(const float* __restrict__ gsum, const float* __restrict__ gcnt,
                        const float* __restrict__ lb, float* __restrict__ outp, int G) {
    int g = blockIdx.x * blockDim.x + threadIdx.x;
    if (g >= G) return;
    float c = gcnt[g];
    if (c < 1.f) c = 1.f;
    outp[g] = gsum[g] / c + lb[0];
}

// ---------------------------------------------------------------------------
extern "C" void kernel_launch(void* const* d_in, const int* in_sizes, int n_in,
                              void* d_out, int out_size, void* d_ws, size_t ws_size,
                              hipStream_t stream) {
    (void)in_sizes; (void)n_in; (void)out_size; (void)ws_size;
    const int N = N_NODES, E = N_EDGES, G = N_GRAPHS;

    const int* X    = (const int*)d_in[0];
    const int* eidx = (const int*)d_in[1];
    const int* src  = eidx;
    const int* dst  = eidx + E;
    const int* batch = (const int*)d_in[2];
    const int* Ed   = (const int*)d_in[3];
    const float* node_tab = (const float*)d_in[4];
    const float* edge_tab = (const float*)d_in[5];
    const float* Wm[3]  = {(const float*)d_in[6],  (const float*)d_in[12], (const float*)d_in[18]};
    const float* asv[3] = {(const float*)d_in[7],  (const float*)d_in[13], (const float*)d_in[19]};
    const float* adv[3] = {(const float*)d_in[8],  (const float*)d_in[14], (const float*)d_in[20]};
    const float* Wev[3] = {(const float*)d_in[9],  (const float*)d_in[15], (const float*)d_in[21]};
    const float* aev[3] = {(const float*)d_in[10], (const float*)d_in[16], (const float*)d_in[22]};
    const float* bbv[3] = {(const float*)d_in[11], (const float*)d_in[17], (const float*)d_in[23]};
    const float* lin_w = (const float*)d_in[24];
    const float* lin_b = (const float*)d_in[25];
    float* outp = (float*)d_out;

    // workspace carve-up
    float* ws = (float*)d_ws;
    size_t off = 0;
    auto alloc = [&](size_t n) { float* p = ws + off; off += n; return p; };
    float* h       = alloc((size_t)N * 16);
    float* xw      = alloc((size_t)N * 32);
    float* out1    = alloc((size_t)N * 32);
    float* out2    = alloc((size_t)N * 32);
    float* out3    = alloc((size_t)N * 32);
    float* ea      = alloc((size_t)E * 2);
    float* edgeval = alloc((size_t)E);
    float* a_s     = alloc(N);
    float* a_d     = alloc(N);
    float* lval    = alloc(N);
    float* maxbuf  = alloc(N);
    float* den     = alloc(N);
    // contiguous block zeroed once per call:
    float* cnt     = alloc(N);
    float* easum   = alloc((size_t)N * 2);
    float* gsum    = alloc(G);
    float* gcnt    = alloc(G);
    float* aecoef  = alloc(8);

    const int B = 256;
    int zn = N * 3 + 2 * G;
    k_zero<<<(zn + B - 1) / B, B, 0, stream>>>(cnt, zn);

    k_encode_nodes<<<((size_t)N * 16 + B - 1) / B, B, 0, stream>>>(X, node_tab, h, N);
    k_encode_edges<<<(E + B - 1) / B, B, 0, stream>>>(Ed, edge_tab, ea, E);
    k_edge_stats<<<(E + B - 1) / B, B, 0, stream>>>(dst, ea, cnt, easum, E);
    for (int l = 0; l < 3; ++l)
        k_aecoef<<<1, 32, 0, stream>>>(Wev[l], aev[l], aecoef + 2 * l);

    const float* xin[3] = {h, out1, out2};
    float* outl[3] = {out1, out2, out3};
    const int Dm[3] = {16, 32, 32};

    for (int l = 0; l < 3; ++l) {
        k_gemm_wmma<<<(N + 127) / 128, B, 0, stream>>>(xin[l], Wm[l], xw, N, Dm[l]);
        k_node_prep<<<(N + B - 1) / B, B, 0, stream>>>(xw, asv[l], adv[l], cnt, easum,
                                                       aecoef + 2 * l, bbv[l], a_s, a_d,
                                                       lval, maxbuf, den, outl[l], N);
        k_edge_alpha<<<(E + B - 1) / B, B, 0, stream>>>(src, dst, ea, aecoef + 2 * l,
                                                        a_s, a_d, edgeval, maxbuf, E);
        k_edge_exp<<<(E + N + B - 1) / B, B, 0, stream>>>(dst, edgeval, maxbuf, den, lval, E, N);
        long long aggT = ((long long)E + N) * 32;
        k_aggregate<<<(unsigned)((aggT + B - 1) / B), B, 0, stream>>>(src, dst, edgeval, den,
                                                                      xw, outl[l], lval, E, N);
    }

    k_pool<<<(N + B - 1) / B, B, 0, stream>>>(out1, out2, out3, lin_w, batch, gsum, gcnt, N);
    k_final<<<(G + B - 1) / B, B, 0, stream>>>(gsum, gcnt, lin_b, outp, G);
}